// TpsGridGen_4896262717696
// MI455X (gfx1250) — compile-verified
//
#include <hip/hip_runtime.h>

#define OUT_H 256
#define OUT_W 192
#define GRID_SZ 5
#define NCP 25               // TPS control points
#define KDIM 28              // 25 RBF + 3 affine, = 7 * 4 (WMMA f32 K-step)
#define BATCH 64
#define HW (OUT_H * OUT_W)   // 49152 = 3072 tiles of 16 pixels

typedef __attribute__((ext_vector_type(2))) float v2f;
typedef __attribute__((ext_vector_type(8))) float v8f;

// control point coordinates: _PY,_PX = np.meshgrid(ax, ax) ('xy') =>
// PX[n] = ax[n/5], PY[n] = ax[n%5], ax = linspace(-1,1,5)
__device__ __forceinline__ float ctrl_px(int n) { return -1.0f + 0.5f * (float)(n / GRID_SZ); }
__device__ __forceinline__ float ctrl_py(int n) { return -1.0f + 0.5f * (float)(n % GRID_SZ); }

// -------------------------------------------------------------------------
// Kernel 1: build L (28x28 TPS system matrix) and invert with Gauss-Jordan
// + partial pivoting (K-block diagonal is zero, pivoting required).
// Single thread; ~28^3 ops, cost irrelevant.
// -------------------------------------------------------------------------
__global__ void tps_build_li(float* __restrict__ Li) {
  if (threadIdx.x != 0 || blockIdx.x != 0) return;
  float A[KDIM][2 * KDIM];
  for (int i = 0; i < KDIM; ++i) {
    for (int j = 0; j < KDIM; ++j) {
      float v = 0.0f;
      if (i < NCP && j < NCP) {
        float dx = ctrl_px(i) - ctrl_px(j);
        float dy = ctrl_py(i) - ctrl_py(j);
        float d2 = dx * dx + dy * dy;
        if (d2 == 0.0f) d2 = 1.0f;        // -> U = 0 on diagonal
        v = d2 * __logf(d2);
      } else if (i < NCP) {               // P block: [1, PX, PY]
        int a = j - NCP;
        v = (a == 0) ? 1.0f : ((a == 1) ? ctrl_px(i) : ctrl_py(i));
      } else if (j < NCP) {               // P^T block
        int a = i - NCP;
        v = (a == 0) ? 1.0f : ((a == 1) ? ctrl_px(j) : ctrl_py(j));
      }
      A[i][j] = v;
      A[i][j + KDIM] = (i == j) ? 1.0f : 0.0f;
    }
  }
  // Gauss-Jordan with partial pivoting
  for (int c = 0; c < KDIM; ++c) {
    int piv = c;
    float best = fabsf(A[c][c]);
    for (int r = c + 1; r < KDIM; ++r) {
      float m = fabsf(A[r][c]);
      if (m > best) { best = m; piv = r; }
    }
    if (piv != c) {
      for (int j = 0; j < 2 * KDIM; ++j) {
        float t = A[c][j]; A[c][j] = A[piv][j]; A[piv][j] = t;
      }
    }
    float inv = 1.0f / A[c][c];
    for (int j = 0; j < 2 * KDIM; ++j) A[c][j] *= inv;
    for (int r = 0; r < KDIM; ++r) {
      if (r == c) continue;
      float f = A[r][c];
      if (f != 0.0f)
        for (int j = 0; j < 2 * KDIM; ++j) A[r][j] -= f * A[c][j];
    }
  }
  for (int i = 0; i < KDIM; ++i)
    for (int j = 0; j < KDIM; ++j)
      Li[i * KDIM + j] = A[i][j + KDIM];
}

// -------------------------------------------------------------------------
// Kernel 2: augmented weights  W'[c][b][k] = sum_{j<25} Li[k][j] * Q_c[b][j]
//   k <  25 : TPS RBF weights (rows of LI_W)
//   k >= 25 : affine coefficients (rows of LI_A)
// theta: [B, 2*NCP], x offsets then y offsets.
// -------------------------------------------------------------------------
__global__ void tps_weights(const float* __restrict__ theta,
                            const float* __restrict__ Li,
                            float* __restrict__ Wp) {
  int idx = blockIdx.x * blockDim.x + threadIdx.x;
  if (idx >= 2 * BATCH * KDIM) return;
  int k = idx % KDIM;
  int b = (idx / KDIM) % BATCH;
  int c = idx / (KDIM * BATCH);   // 0 = X, 1 = Y
  float s = 0.0f;
  for (int j = 0; j < NCP; ++j) {
    float base = (c == 0) ? ctrl_px(j) : ctrl_py(j);
    float q = theta[b * (2 * NCP) + c * NCP + j] + base;
    s += Li[k * KDIM + j] * q;
  }
  Wp[(c * BATCH + b) * KDIM + k] = s;
}

// -------------------------------------------------------------------------
// Kernel 3: the GEMM  out[c][b][p] = sum_k W'[c][b][k] * U'[p][k]
// via V_WMMA_F32_16X16X4_F32.  M = batch (16), N = pixel (16), K = 28 = 7x4.
// One wave per 16-pixel tile; 8 waves/block; 3072 tiles total.
// U'[p][25]=1, U'[p][26]=gx, U'[p][27]=gy folds the affine term in.
// -------------------------------------------------------------------------
__device__ __forceinline__ float ueval(int k, float gx, float gy) {
  if (k >= NCP) {
    if (k == NCP) return 1.0f;
    return (k == NCP + 1) ? gx : gy;
  }
  float dx = gx - ctrl_px(k);
  float dy = gy - ctrl_py(k);
  float d2 = dx * dx + dy * dy;
  if (d2 == 0.0f) d2 = 1.0f;
  return d2 * __logf(d2);
}

__global__ __launch_bounds__(256) void tps_main(const float* __restrict__ Wp,
                                                float* __restrict__ out) {
  const int lane = threadIdx.x & 31;
  const int wave = threadIdx.x >> 5;
  const int tile = blockIdx.x * 8 + wave;   // 16 consecutive pixels
  const int p0 = tile * 16;

  // This lane's pixel for the B (U) operand columns and for stores (N = lane&15)
  const int myp = p0 + (lane & 15);
  const int h = myp / OUT_W;
  const int w = myp - h * OUT_W;
  const float gx = -1.0f + 2.0f * (float)w / (float)(OUT_W - 1);
  const float gy = -1.0f + 2.0f * (float)h / (float)(OUT_H - 1);

  const float* __restrict__ WpX = Wp;
  const float* __restrict__ WpY = Wp + BATCH * KDIM;

  const v8f vzero = {0.f, 0.f, 0.f, 0.f, 0.f, 0.f, 0.f, 0.f};
  v8f accX[4] = {vzero, vzero, vzero, vzero};   // 4 batch tiles, X coord
  v8f accY[4] = {vzero, vzero, vzero, vzero};   // 4 batch tiles, Y coord

  // 32-bit operand striping (ISA 7.12.2): VGPR v, lane<16 -> K = kb+v,
  // lane>=16 -> K = kb+2+v.  Same pattern used for A (M=lane&15 rows) and B.
  const int khalf = (lane < 16) ? 0 : 2;
  const int arow = lane & 15;   // batch-within-tile for A operand

#pragma unroll
  for (int kblk = 0; kblk < 7; ++kblk) {
    const int kb = kblk * 4 + khalf;
    v2f bfrag;
    bfrag.x = ueval(kb, gx, gy);
    bfrag.y = ueval(kb + 1, gx, gy);
#pragma unroll
    for (int m = 0; m < 4; ++m) {
      const int b = m * 16 + arow;
      v2f ax = *(const v2f*)(WpX + b * KDIM + kb);
      v2f ay = *(const v2f*)(WpY + b * KDIM + kb);
      accX[m] = __builtin_amdgcn_wmma_f32_16x16x4_f32(
          false, ax, false, bfrag, (short)0, accX[m], false, false);
      accY[m] = __builtin_amdgcn_wmma_f32_16x16x4_f32(
          false, ay, false, bfrag, (short)0, accY[m], false, false);
    }
  }

  // D layout: VGPR v -> M = v (lanes 0-15) / v+8 (lanes 16-31), N = lane&15.
  // Store interleaved {x,y} pairs: each b64 store = two contiguous 128B runs.
  const int mofs = (lane < 16) ? 0 : 8;
#pragma unroll
  for (int m = 0; m < 4; ++m) {
#pragma unroll
    for (int v = 0; v < 8; ++v) {
      const int b = m * 16 + v + mofs;
      v2f st;
      st.x = accX[m][v];
      st.y = accY[m][v];
      *(v2f*)(out + ((size_t)b * HW + (size_t)myp) * 2) = st;
    }
  }
}

// -------------------------------------------------------------------------
extern "C" void kernel_launch(void* const* d_in, const int* in_sizes, int n_in,
                              void* d_out, int out_size, void* d_ws, size_t ws_size,
                              hipStream_t stream) {
  const float* theta = (const float*)d_in[0];   // [64, 50] f32
  float* out = (float*)d_out;                   // [64, 256, 192, 2] f32
  float* ws = (float*)d_ws;
  float* Li = ws;                               // 28*28 floats
  float* Wp = ws + KDIM * KDIM;                 // 2*64*28 floats

  tps_build_li<<<1, 32, 0, stream>>>(Li);
  tps_weights<<<(2 * BATCH * KDIM + 255) / 256, 256, 0, stream>>>(theta, Li, Wp);
  tps_main<<<(HW / 16) / 8, 256, 0, stream>>>(Wp, out);
}